// DirSageConv_31610959299133
// MI455X (gfx1250) — compile-verified
//
#include <hip/hip_runtime.h>

// ---------------------------------------------------------------------------
// DirSageConv for MI455X (gfx1250, wave32)
//   Phase 1: zero workspace accumulators
//   Phase 2: edge scatter (both directions) with hardware global_atomic_add_f32
//            (all buffers L2-resident: x=5.1MB, each accumulator=5.1MB, L2=192MB)
//   Phase 3: fused 3-way GEMM via V_WMMA_F32_16X16X4_F32, mean/alpha folded in
// ---------------------------------------------------------------------------

#define N_NODES 10000
#define N_EDGES 640000
#define D_FEAT  128
static constexpr float ALPHA = 0.5f;

typedef __attribute__((ext_vector_type(2))) float v2f;
typedef __attribute__((ext_vector_type(8))) float v8f;

// Guaranteed single-instruction fp32 atomic (GLOBAL_ATOMIC_ADD_F32, no return).
__device__ __forceinline__ void atomic_add_f32(float* p, float v) {
    unsafeAtomicAdd(p, v);   // HIP: lowers to global_atomic_add_f32 (no CAS loop)
}

// ---------------- Phase 1: zero the aggregation workspace -------------------
__global__ __launch_bounds__(256) void zero_ws_kernel(float* __restrict__ p, int n) {
    int i = blockIdx.x * blockDim.x + threadIdx.x;
    if (i < n) p[i] = 0.0f;
}

// ---------------- Phase 2: bidirectional edge scatter -----------------------
// One wave32 per edge: lane handles 4 consecutive features (float4 read of x,
// 4 global_atomic_add_f32 per direction). Lanes 0/1 bump the degree counters.
__global__ __launch_bounds__(256) void scatter_kernel(
    const float* __restrict__ x,
    const long long* __restrict__ src_idx,
    const long long* __restrict__ dst_idx,
    float* __restrict__ agg_s2d,   // accumulates x[src] at dst
    float* __restrict__ agg_d2s,   // accumulates x[dst] at src
    float* __restrict__ cnt_dst,
    float* __restrict__ cnt_src)
{
    const int t    = blockIdx.x * blockDim.x + threadIdx.x;
    const int e    = t >> 5;
    const int lane = t & 31;
    if (e >= N_EDGES) return;

    // prefetch edge indices ahead (global_prefetch_b8)
    if (e + 64 < N_EDGES) {
        __builtin_prefetch(&src_idx[e + 64], 0, 1);
        __builtin_prefetch(&dst_idx[e + 64], 0, 1);
    }

    const int s = (int)src_idx[e];
    const int d = (int)dst_idx[e];

    const float4 xs = *(const float4*)(x + (size_t)s * D_FEAT + lane * 4);
    const float4 xd = *(const float4*)(x + (size_t)d * D_FEAT + lane * 4);

    float* ps = agg_s2d + (size_t)d * D_FEAT + lane * 4;   // x[src] -> dst bucket
    float* pd = agg_d2s + (size_t)s * D_FEAT + lane * 4;   // x[dst] -> src bucket

    atomic_add_f32(ps + 0, xs.x); atomic_add_f32(ps + 1, xs.y);
    atomic_add_f32(ps + 2, xs.z); atomic_add_f32(ps + 3, xs.w);
    atomic_add_f32(pd + 0, xd.x); atomic_add_f32(pd + 1, xd.y);
    atomic_add_f32(pd + 2, xd.z); atomic_add_f32(pd + 3, xd.w);

    if (lane == 0) atomic_add_f32(cnt_dst + d, 1.0f);
    if (lane == 1) atomic_add_f32(cnt_src + s, 1.0f);
}

// ---------------- Phase 3: fused triple GEMM via fp32 WMMA ------------------
// out = x@W_self + (1-a)*(mean_s2d@W_s2d) + a*(mean_d2s@W_d2s) + combined bias
// One wave -> one 16x16 tile. Block = 256 threads = 8 waves = full 128 cols.
// Grid = 10000/16 = 625 row strips (exact, EXEC all-1s for WMMA).
__global__ __launch_bounds__(256) void gemm_kernel(
    const float* __restrict__ x,
    const float* __restrict__ agg_s2d, const float* __restrict__ cnt_dst,
    const float* __restrict__ agg_d2s, const float* __restrict__ cnt_src,
    const float* __restrict__ W_self, const float* __restrict__ b_self,
    const float* __restrict__ W_s2d,  const float* __restrict__ b_s2d,
    const float* __restrict__ W_d2s,  const float* __restrict__ b_d2s,
    float* __restrict__ out)
{
    const int lane = threadIdx.x & 31;
    const int wave = threadIdx.x >> 5;
    const int row0 = blockIdx.x * 16;     // M tile base
    const int col0 = wave * 16;           // N tile base
    const int half = lane >> 4;           // 0: lanes 0-15, 1: lanes 16-31
    const int l15  = lane & 15;

    // A-fragment row for this lane (M = l15), K sub-offset 2*half (ISA A layout)
    const int  arow = row0 + l15;
    const int  koff = 2 * half;

    v8f acc = {};

    #pragma unroll
    for (int m = 0; m < 3; ++m) {
        const float* A; const float* W; float rscale;
        if (m == 0)      { A = x;       W = W_self; rscale = 1.0f; }
        else if (m == 1) { A = agg_s2d; W = W_s2d;
                           rscale = (1.0f - ALPHA) / fmaxf(cnt_dst[arow], 1.0f); }
        else             { A = agg_d2s; W = W_d2s;
                           rscale = ALPHA / fmaxf(cnt_src[arow], 1.0f); }

        const float* ap = A + (size_t)arow * D_FEAT + koff;   // 8B aligned
        #pragma unroll 4
        for (int k0 = 0; k0 < D_FEAT; k0 += 4) {
            v2f a = *(const v2f*)(ap + k0);                   // K = k0+koff, k0+koff+1
            a *= rscale;                                      // fold mean + alpha scale
            v2f b;
            b[0] = W[(size_t)(k0 + koff + 0) * D_FEAT + col0 + l15];
            b[1] = W[(size_t)(k0 + koff + 1) * D_FEAT + col0 + l15];
            acc = __builtin_amdgcn_wmma_f32_16x16x4_f32(
                      /*neg_a=*/false, a, /*neg_b=*/false, b,
                      /*c_mod=*/(short)0, acc,
                      /*reuse_a=*/false, /*reuse_b=*/false);
        }
    }

    // Combined bias for this lane's column, added to all 8 accumulator rows.
    const int col = col0 + l15;
    const float bias = b_self[col] + (1.0f - ALPHA) * b_s2d[col] + ALPHA * b_d2s[col];

    // C/D layout: VGPR r -> row r (lanes 0-15) / row r+8 (lanes 16-31)
    #pragma unroll
    for (int r = 0; r < 8; ++r) {
        const int rr = row0 + r + 8 * half;
        out[(size_t)rr * D_FEAT + col] = acc[r] + bias;
    }
}

// ---------------------------------------------------------------------------
extern "C" void kernel_launch(void* const* d_in, const int* in_sizes, int n_in,
                              void* d_out, int out_size, void* d_ws, size_t ws_size,
                              hipStream_t stream) {
    // setup_inputs order: x, edge_index, W_s2d, b_s2d, W_d2s, b_d2s, W_self, b_self
    const float*     x      = (const float*)d_in[0];
    const long long* eidx   = (const long long*)d_in[1];   // int64 [2, N_EDGES]
    const float*     W_s2d  = (const float*)d_in[2];
    const float*     b_s2d  = (const float*)d_in[3];
    const float*     W_d2s  = (const float*)d_in[4];
    const float*     b_d2s  = (const float*)d_in[5];
    const float*     W_self = (const float*)d_in[6];
    const float*     b_self = (const float*)d_in[7];
    float*           out    = (float*)d_out;

    const long long* src_idx = eidx;
    const long long* dst_idx = eidx + N_EDGES;

    // Workspace: agg_s2d [N,D] | agg_d2s [N,D] | cnt_dst [N] | cnt_src [N]
    float* agg_s2d = (float*)d_ws;
    float* agg_d2s = agg_s2d + (size_t)N_NODES * D_FEAT;
    float* cnt_dst = agg_d2s + (size_t)N_NODES * D_FEAT;
    float* cnt_src = cnt_dst + N_NODES;

    const int n_zero = 2 * N_NODES * D_FEAT + 2 * N_NODES;
    zero_ws_kernel<<<(n_zero + 255) / 256, 256, 0, stream>>>(agg_s2d, n_zero);

    const long long scat_threads = (long long)N_EDGES * 32;
    scatter_kernel<<<(int)((scat_threads + 255) / 256), 256, 0, stream>>>(
        x, src_idx, dst_idx, agg_s2d, agg_d2s, cnt_dst, cnt_src);

    gemm_kernel<<<N_NODES / 16, 256, 0, stream>>>(
        x, agg_s2d, cnt_dst, agg_d2s, cnt_src,
        W_self, b_self, W_s2d, b_s2d, W_d2s, b_d2s, out);
}